// GCN_18073222382223
// MI455X (gfx1250) — compile-verified
//
#include <hip/hip_runtime.h>
#include <hip/hip_bf16.h>

#define D 128
#define EPSN 1e-5f
#define EPW 4          // edges per wave in the scatter kernel
#define GW  8          // waves per GEMM workgroup

typedef float v2f __attribute__((ext_vector_type(2)));
typedef float v8f __attribute__((ext_vector_type(8)));

// ---------------------------------------------------------------- utilities
__global__ void k_fill(float* __restrict__ p, int n, float v) {
    int i = blockIdx.x * blockDim.x + threadIdx.x;
    if (i < n) p[i] = v;
}

__global__ void k_deg(const int* __restrict__ col, float* __restrict__ deg, int E) {
    int i = blockIdx.x * blockDim.x + threadIdx.x;
    if (i < E) atomicAdd(&deg[col[i]], 1.0f);
}

__global__ void k_rsqrt(float* __restrict__ p, int n) {
    int i = blockIdx.x * blockDim.x + threadIdx.x;
    if (i < n) p[i] = rsqrtf(p[i]);   // deg >= 1 always (self-loops)
}

// ---------------------------------------------------------------- WMMA GEMM
// C[M x 128] = A[M x 128] * W[128 x 128]  (+bias, +relu optional)
// 8 waves per block; W staged once per block into LDS via gfx1250 async
// load-to-LDS (fallback: plain copy). Each wave owns a 16-row slab and
// accumulates 8 x V_WMMA_F32_16X16X4_F32 tiles covering all 128 columns.
template<bool BIAS, bool RELU>
__global__ __launch_bounds__(256) void
k_gemm128(const float* __restrict__ A, const float* __restrict__ W,
          const float* __restrict__ bias, float* __restrict__ C, int M) {
    __shared__ float lds_w[D * D];     // 64 KB (of 320 KB/WGP)
    const int tid = threadIdx.x;

#if defined(__AMDGCN__) && __has_builtin(__builtin_amdgcn_global_load_async_to_lds_b128) && \
    __has_builtin(__builtin_amdgcn_s_wait_asynccnt)
    {
        typedef int v4i_ __attribute__((vector_size(16)));
        typedef __attribute__((address_space(1))) v4i_ gv4i;
        typedef __attribute__((address_space(3))) v4i_ lv4i;
        gv4i* gsrc = (gv4i*)W;         // global source, 16B units
        lv4i* ldst = (lv4i*)lds_w;     // LDS destination, 16B units
        #pragma unroll
        for (int i = 0; i < (D * D) / (4 * 256); ++i) {   // 4096 vec4 / 256 thr = 16
            const int idx = tid + i * 256;
            __builtin_amdgcn_global_load_async_to_lds_b128(gsrc + idx, ldst + idx, 0, 0);
        }
        __builtin_amdgcn_s_wait_asynccnt(0);
    }
#else
    for (int i = tid * 4; i < D * D; i += 256 * 4) {
        float4 t = *(const float4*)(W + i);
        *(float4*)(&lds_w[i]) = t;
    }
#endif
    __syncthreads();

    const int lane = tid & 31;
    const int hi   = lane >> 4;            // 0: lanes 0-15, 1: lanes 16-31
    const int mn   = lane & 15;
    const int slab = blockIdx.x * GW + (tid >> 5);
    const int row0 = slab * 16;
    if (row0 >= M) return;                 // wave-uniform: EXEC stays all-1s

    const float* __restrict__ arow = A + (size_t)(row0 + mn) * D;

    v8f acc[8] = {};

    for (int k = 0; k < D; k += 4) {
        // A 16x4 fragment: VGPR0 = K {0|2}, VGPR1 = K {1|3} by lane half
        v2f a = *(const v2f*)(arow + k + 2 * hi);
        const float* wr = &lds_w[(k + 2 * hi) * D];   // row K, next row = +D
        #pragma unroll
        for (int t = 0; t < 8; ++t) {
            const int n = t * 16 + mn;
            v2f b;
            b.x = wr[n];           // K = k+2*hi
            b.y = wr[D + n];       // K = k+2*hi+1
            acc[t] = __builtin_amdgcn_wmma_f32_16x16x4_f32(
                false, a, false, b, (short)0, acc[t], false, false);
        }
    }

    #pragma unroll
    for (int t = 0; t < 8; ++t) {
        const int cc = t * 16 + mn;
        #pragma unroll
        for (int r = 0; r < 8; ++r) {
            const int rr = row0 + r + 8 * hi;  // C layout: VGPR r -> row r (+8 hi)
            float v = acc[t][r];
            if (BIAS) v += bias[cc];
            if (RELU) v = fmaxf(v, 0.0f);
            C[(size_t)rr * D + cc] = v;
        }
    }
}

// ---------------------------------------------------------------- GCN aggregation
// out[i,:] = h[i,:]*dinv[i]^2 + bias  (self-loop term + bias, also zero-inits)
__global__ void k_self(const float* __restrict__ h, const float* __restrict__ dinv,
                       const float* __restrict__ bias, float* __restrict__ out, int total) {
    int i = blockIdx.x * blockDim.x + threadIdx.x;
    if (i >= total) return;
    int n = i >> 7, c = i & 127;
    float di = dinv[n];
    out[i] = h[i] * di * di + bias[c];
}

// one wave per EPW edges: read indices up-front, prefetch the 512B source
// rows (global_prefetch_b8), then float4 coalesced gather + scaled
// global_atomic_add_f32 scatter. All control flow is wave-uniform.
__global__ __launch_bounds__(256) void
k_edge(const float* __restrict__ h, const int* __restrict__ row,
       const int* __restrict__ col, const float* __restrict__ dinv,
       float* __restrict__ out, int E) {
    const int lane = threadIdx.x & 31;
    const int wid  = (blockIdx.x << 3) + (threadIdx.x >> 5);
    const int e0   = wid * EPW;
    if (e0 >= E) return;
    const int ne = min(EPW, E - e0);

    int r[EPW], c[EPW];
    #pragma unroll
    for (int j = 0; j < EPW; ++j) {
        if (j < ne) {
            r[j] = row[e0 + j];
            c[j] = col[e0 + j];
            __builtin_prefetch(h + (size_t)r[j] * D + lane * 4, 0, 3);
        }
    }
    #pragma unroll
    for (int j = 0; j < EPW; ++j) {
        if (j < ne) {
            float nrm = dinv[r[j]] * dinv[c[j]];
            float4 v = ((const float4*)(h + (size_t)r[j] * D))[lane];
            float* op = out + (size_t)c[j] * D + lane * 4;
            atomicAdd(op + 0, v.x * nrm);
            atomicAdd(op + 1, v.y * nrm);
            atomicAdd(op + 2, v.z * nrm);
            atomicAdd(op + 3, v.w * nrm);
        }
    }
}

// ---------------------------------------------------------------- GraphNorm
// 128 threads per block, thread = channel -> fully coalesced row sweeps.
__global__ void k_colsum(const float* __restrict__ x, float* __restrict__ sum, int n) {
    int c = threadIdx.x;
    int rows_per = (n + gridDim.x - 1) / gridDim.x;
    int r0 = blockIdx.x * rows_per;
    int r1 = min(n, r0 + rows_per);
    float s = 0.0f;
    for (int r = r0; r < r1; ++r) s += x[(size_t)r * D + c];
    atomicAdd(&sum[c], s);
}

__global__ void k_colvar(const float* __restrict__ x, const float* __restrict__ sum,
                         const float* __restrict__ alpha, float* __restrict__ sumsq, int n) {
    int c = threadIdx.x;
    float mu = alpha[c] * (sum[c] / (float)n);
    int rows_per = (n + gridDim.x - 1) / gridDim.x;
    int r0 = blockIdx.x * rows_per;
    int r1 = min(n, r0 + rows_per);
    float s = 0.0f;
    for (int r = r0; r < r1; ++r) {
        float d = x[(size_t)r * D + c] - mu;
        s += d * d;
    }
    atomicAdd(&sumsq[c], s);
}

__global__ void k_norm_relu(const float* __restrict__ x, const float* __restrict__ sum,
                            const float* __restrict__ sumsq, const float* __restrict__ w,
                            const float* __restrict__ b, const float* __restrict__ alpha,
                            float* __restrict__ out, int n, int total) {
    int i = blockIdx.x * blockDim.x + threadIdx.x;
    if (i >= total) return;
    int c = i & 127;
    float mu  = alpha[c] * (sum[c] / (float)n);
    float var = sumsq[c] / (float)n;
    float v = w[c] * (x[i] - mu) * rsqrtf(var + EPSN) + b[c];
    out[i] = fmaxf(v, 0.0f);
}

// ---------------------------------------------------------------- final head
// out[r] = dot(y[r,:], w) + b   -- one wave per row, shuffle reduction (wave32)
__global__ void k_head(const float* __restrict__ y, const float* __restrict__ w,
                       const float* __restrict__ b, float* __restrict__ out, int n) {
    int r = (blockIdx.x << 3) + (threadIdx.x >> 5);
    if (r >= n) return;
    int lane = threadIdx.x & 31;
    float4 v  = ((const float4*)(y + (size_t)r * D))[lane];
    float4 wv = ((const float4*)w)[lane];
    float s = v.x * wv.x + v.y * wv.y + v.z * wv.z + v.w * wv.w;
    #pragma unroll
    for (int off = 16; off > 0; off >>= 1) s += __shfl_xor(s, off, 32);
    if (lane == 0) out[r] = s + b[0];
}

// ---------------------------------------------------------------- launcher
extern "C" void kernel_launch(void* const* d_in, const int* in_sizes, int n_in,
                              void* d_out, int out_size, void* d_ws, size_t ws_size,
                              hipStream_t stream) {
    const float* x   = (const float*)d_in[0];
    const int*   ei  = (const int*)d_in[1];
    const int E = in_sizes[1] / 2;
    const int* row = ei;          // edge_index[0]
    const int* col = ei + E;      // edge_index[1]
    const float* W0  = (const float*)d_in[2];
    const float* b0  = (const float*)d_in[3];
    const float* W1  = (const float*)d_in[4];
    const float* b1  = (const float*)d_in[5];
    const float* g0w = (const float*)d_in[6];
    const float* g0b = (const float*)d_in[7];
    const float* g0a = (const float*)d_in[8];
    const float* g1w = (const float*)d_in[9];
    const float* g1b = (const float*)d_in[10];
    const float* g1a = (const float*)d_in[11];
    const float* l0w = (const float*)d_in[12];
    const float* l0b = (const float*)d_in[13];
    const float* l1w = (const float*)d_in[14];
    const float* l1b = (const float*)d_in[15];
    float* outp = (float*)d_out;

    const int N = in_sizes[0] / D;     // 100000
    const int total = N * D;

    // workspace carve-up (512B aligned blocks)
    char* ws = (char*)d_ws;
    size_t off = 0;
    float* dinv  = (float*)(ws + off); off += (((size_t)N * 4 + 511) & ~(size_t)511);
    float* stats = (float*)(ws + off); off += 1024;                 // 256 floats: sum | sumsq
    float* bufA  = (float*)(ws + off); off += (size_t)total * 4;
    float* bufB  = (float*)(ws + off);

    const dim3 B256(256);
    const int gN   = (N + 255) / 256;
    const int gT   = (total + 255) / 256;
    const int gE   = (E + 255) / 256;
    const int gEw  = (E + 8 * EPW - 1) / (8 * EPW);   // EPW edges per wave
    const int gRow = (N / 16 + GW - 1) / GW;          // GEMM: GW slabs per block
    const int gHd  = (N + 7) / 8;

    // degree -> dinv = rsqrt(deg), deg initialized to 1 for self-loops
    k_fill<<<gN, B256, 0, stream>>>(dinv, N, 1.0f);
    k_deg<<<gE, B256, 0, stream>>>(col, dinv, E);
    k_rsqrt<<<gN, B256, 0, stream>>>(dinv, N);

    // ---- layer 0: h = x @ W0 ; agg = A_hat h + b0 ; x1 = relu(norm(agg))
    k_gemm128<false, false><<<gRow, B256, 0, stream>>>(x, W0, nullptr, bufA, N);
    k_self<<<gT, B256, 0, stream>>>(bufA, dinv, b0, bufB, total);
    k_edge<<<gEw, B256, 0, stream>>>(bufA, row, col, dinv, bufB, E);
    k_fill<<<1, B256, 0, stream>>>(stats, 256, 0.0f);
    k_colsum<<<1024, 128, 0, stream>>>(bufB, stats, N);
    k_colvar<<<1024, 128, 0, stream>>>(bufB, stats, g0a, stats + 128, N);
    k_norm_relu<<<gT, B256, 0, stream>>>(bufB, stats, stats + 128, g0w, g0b, g0a, bufA, N, total);

    // ---- layer 1
    k_gemm128<false, false><<<gRow, B256, 0, stream>>>(bufA, W1, nullptr, bufB, N);
    k_self<<<gT, B256, 0, stream>>>(bufB, dinv, b1, bufA, total);
    k_edge<<<gEw, B256, 0, stream>>>(bufB, row, col, dinv, bufA, E);
    k_fill<<<1, B256, 0, stream>>>(stats, 256, 0.0f);
    k_colsum<<<1024, 128, 0, stream>>>(bufA, stats, N);
    k_colvar<<<1024, 128, 0, stream>>>(bufA, stats, g1a, stats + 128, N);
    k_norm_relu<<<gT, B256, 0, stream>>>(bufA, stats, stats + 128, g1w, g1b, g1a, bufB, N, total);

    // ---- MLP head
    k_gemm128<true, true><<<gRow, B256, 0, stream>>>(bufB, l0w, l0b, bufA, N);
    k_head<<<gHd, B256, 0, stream>>>(bufA, l1w, l1b, outp, N);
}